// GCNEncoder_42640435314994
// MI455X (gfx1250) — compile-verified
//
#include <hip/hip_runtime.h>

// ---------------------------------------------------------------------------
// GCN encoder for MI455X (gfx1250, wave32).
//   layer: hW = h @ W (fp32 WMMA 16x16x4), agg = scatter(norm * hW[src] -> dst)
//          + self_coef * hW, out = relu(agg + b)
// GEMM flops are negligible (6.6 GFLOP); hW (51MB) and out (102MB) are
// L2-resident (192MB), so the scatter phase is L2-atomic bound. fp32 WMMA
// keeps reference precision at no cost since matrix units are not the limit.
// ---------------------------------------------------------------------------

typedef float v2f __attribute__((ext_vector_type(2)));
typedef float v8f __attribute__((ext_vector_type(8)));

#define D_DIM 128
#define H_DIM 128
#define OUT_LD 256

__device__ __forceinline__ void atomicAddF32(float* p, float v) {
  // lowers to global_atomic_add_f32 (non-returning, relaxed, device scope)
  __hip_atomic_fetch_add(p, v, __ATOMIC_RELAXED, __HIP_MEMORY_SCOPE_AGENT);
}

// ---- degree / norm precompute ---------------------------------------------

__global__ void k_deg_init(float* __restrict__ deg, int N) {
  int i = blockIdx.x * blockDim.x + threadIdx.x;
  if (i < N) deg[i] = 1.0f;  // self-loop weight
}

__global__ void k_deg_scatter(float* __restrict__ deg,
                              const float* __restrict__ w,
                              const long long* __restrict__ dst, int E) {
  int e = blockIdx.x * blockDim.x + threadIdx.x;
  if (e < E) atomicAddF32(&deg[dst[e]], w[e]);
}

__global__ void k_dis(float* __restrict__ deg, int N) {
  int i = blockIdx.x * blockDim.x + threadIdx.x;
  if (i < N) {
    float d = deg[i];
    deg[i] = (d > 0.0f) ? rsqrtf(d) : 0.0f;  // in place: deg -> dis
  }
}

__global__ void k_norm(float* __restrict__ norm,
                       const float* __restrict__ dis,
                       const float* __restrict__ w,
                       const long long* __restrict__ src,
                       const long long* __restrict__ dst, int E) {
  int e = blockIdx.x * blockDim.x + threadIdx.x;
  if (e < E) norm[e] = dis[src[e]] * w[e] * dis[dst[e]];
}

// ---- fp32 WMMA GEMM: C[N,128] = A[N,K=128](lda) * W[128,128] --------------
// One wave per 16x16 output tile, K-loop in steps of 4 using
// V_WMMA_F32_16X16X4_F32. A-frag (16x4): lane m=lane&15 holds row m,
// K = 2*(lane>>4) + {0,1}. B-frag (4x16): lane n=lane&15 holds col n, same K
// split. C/D: VGPR v, lanes<16 -> M=v, lanes>=16 -> M=v+8.

__global__ void k_gemm_wmma(const float* __restrict__ A, int lda,
                            const float* __restrict__ W,
                            float* __restrict__ C, int tiles) {
  int wid  = (blockIdx.x * blockDim.x + threadIdx.x) >> 5;
  int lane = threadIdx.x & 31;
  if (wid >= tiles) return;                 // wave-uniform: EXEC stays all-1s
  int tileM = wid >> 3;                     // H/16 = 8 tiles along columns
  int tileN = wid & 7;
  int row0 = tileM << 4;
  int col0 = tileN << 4;

  int m      = lane & 15;                   // row (A) / col (B) within tile
  int laneHi = lane >> 4;
  int khi    = laneHi << 1;                 // K sub-offset {0,2}

  const float* arow = A + (size_t)(row0 + m) * lda + khi;
  const float* wcol = W + (size_t)khi * H_DIM + col0 + m;

  v8f acc = {};
#pragma unroll
  for (int k0 = 0; k0 < D_DIM; k0 += 4) {
    v2f a = *(const v2f*)(arow + k0);       // A[row, k0+khi .. +1] contiguous
    v2f b;
    b.x = wcol[(size_t)k0 * H_DIM];         // W[k0+khi  , col]
    b.y = wcol[(size_t)k0 * H_DIM + H_DIM]; // W[k0+khi+1, col]
    acc = __builtin_amdgcn_wmma_f32_16x16x4_f32(
        /*neg_a=*/false, a, /*neg_b=*/false, b,
        /*c_mod=*/(short)0, acc, /*reuse_a=*/false, /*reuse_b=*/false);
  }

  float* crow = C + (size_t)(row0 + (laneHi << 3)) * H_DIM + col0 + m;
#pragma unroll
  for (int v = 0; v < 8; ++v) crow[(size_t)v * H_DIM] = acc[v];
}

// ---- self-loop init: out[n, colOff+h] = dis[n]^2 * hW[n,h] ----------------
// Also fully overwrites the (poisoned) output columns before atomics land.

__global__ void k_self_init(float* __restrict__ out, int colOff,
                            const float* __restrict__ hW,
                            const float* __restrict__ dis, int N) {
  int idx = blockIdx.x * blockDim.x + threadIdx.x;
  int n  = idx >> 5;                        // 32 threads/row, float4 each
  int h4 = (idx & 31) << 2;
  if (n < N) {
    float sc = dis[n];
    sc *= sc;                               // self_coef = 1/deg
    float4 v = *(const float4*)(hW + (size_t)n * H_DIM + h4);
    float4 o = {sc * v.x, sc * v.y, sc * v.z, sc * v.w};
    *(float4*)(out + (size_t)n * OUT_LD + colOff + h4) = o;
  }
}

// ---- edge scatter: out[dst, colOff+:] += norm[e] * hW[src, :] -------------
// One wave per edge; lane handles 4 contiguous floats (float4 gather from
// L2-resident hW, 4 fp32 atomics to L2-resident out).

__global__ void k_edge_scatter(float* __restrict__ out, int colOff,
                               const float* __restrict__ hW,
                               const float* __restrict__ norm,
                               const long long* __restrict__ src,
                               const long long* __restrict__ dst, int E) {
  int wid  = (blockIdx.x * blockDim.x + threadIdx.x) >> 5;
  int lane = threadIdx.x & 31;
  if (wid >= E) return;
  long long s = src[wid];
  long long d = dst[wid];
  float nm = norm[wid];
  float4 v = *(const float4*)(hW + (size_t)s * H_DIM + (lane << 2));
  float* o = out + (size_t)d * OUT_LD + colOff + (lane << 2);
  atomicAddF32(o + 0, nm * v.x);
  atomicAddF32(o + 1, nm * v.y);
  atomicAddF32(o + 2, nm * v.z);
  atomicAddF32(o + 3, nm * v.w);
}

// ---- finalize: out = relu(out + b) ----------------------------------------

__global__ void k_finalize(float* __restrict__ out, int colOff,
                           const float* __restrict__ bias, int N) {
  int idx = blockIdx.x * blockDim.x + threadIdx.x;
  int n  = idx >> 5;
  int h4 = (idx & 31) << 2;
  if (n < N) {
    float* p = out + (size_t)n * OUT_LD + colOff + h4;
    float4 v = *(float4*)p;
    float4 b = *(const float4*)(bias + h4);
    v.x = fmaxf(v.x + b.x, 0.0f);
    v.y = fmaxf(v.y + b.y, 0.0f);
    v.z = fmaxf(v.z + b.z, 0.0f);
    v.w = fmaxf(v.w + b.w, 0.0f);
    *(float4*)p = v;
  }
}

static inline size_t align256(size_t x) { return (x + 255) & ~(size_t)255; }

extern "C" void kernel_launch(void* const* d_in, const int* in_sizes, int n_in,
                              void* d_out, int out_size, void* d_ws, size_t ws_size,
                              hipStream_t stream) {
  const float*     x   = (const float*)d_in[0];
  const float*     ew  = (const float*)d_in[1];
  const float*     W0  = (const float*)d_in[2];
  const float*     b0  = (const float*)d_in[3];
  const float*     W1  = (const float*)d_in[4];
  const float*     b1  = (const float*)d_in[5];
  const long long* ei  = (const long long*)d_in[6];  // int64 [2,E]

  const int N = in_sizes[0] / D_DIM;
  const int E = in_sizes[1];
  const long long* srcI = ei;        // edge_index[0]
  const long long* dstI = ei + E;    // edge_index[1]

  // workspace: dis[N] | norm[E] | hW[N*128]   (~58 MB total)
  char*  ws   = (char*)d_ws;
  float* dis  = (float*)ws;
  float* norm = (float*)(ws + align256((size_t)N * 4));
  float* hW   = (float*)(ws + align256((size_t)N * 4) + align256((size_t)E * 4));
  float* out  = (float*)d_out;

  const int T = 256;
  const int gN  = (N + T - 1) / T;
  const int gE  = (E + T - 1) / T;
  const int gNw = ((size_t)N * 32 + T - 1) / T;   // 32 threads per node row
  const int gEw = (E + 7) / 8;                    // 8 waves (edges) per block
  const int tiles  = (N / 16) * (H_DIM / 16);     // N=100000 -> 50000 full tiles
  const int gTiles = (tiles + 7) / 8;

  // norm precompute (shared by both layers)
  k_deg_init   <<<gN, T, 0, stream>>>(dis, N);
  k_deg_scatter<<<gE, T, 0, stream>>>(dis, ew, dstI, E);
  k_dis        <<<gN, T, 0, stream>>>(dis, N);
  k_norm       <<<gE, T, 0, stream>>>(norm, dis, ew, srcI, dstI, E);

  // layer 1: h1 -> out columns [0,128)
  k_gemm_wmma   <<<gTiles, T, 0, stream>>>(x, D_DIM, W0, hW, tiles);
  k_self_init   <<<gNw, T, 0, stream>>>(out, 0, hW, dis, N);
  k_edge_scatter<<<gEw, T, 0, stream>>>(out, 0, hW, norm, srcI, dstI, E);
  k_finalize    <<<gNw, T, 0, stream>>>(out, 0, b0, N);

  // layer 2: A = h1 read from out (lda=256), h2 -> out columns [128,256)
  k_gemm_wmma   <<<gTiles, T, 0, stream>>>(out, OUT_LD, W1, hW, tiles);
  k_self_init   <<<gNw, T, 0, stream>>>(out, H_DIM, hW, dis, N);
  k_edge_scatter<<<gEw, T, 0, stream>>>(out, H_DIM, hW, norm, srcI, dstI, E);
  k_finalize    <<<gNw, T, 0, stream>>>(out, H_DIM, b1, N);
}